// MultiBoxLoss_84765474554227
// MI455X (gfx1250) — compile-verified
//
#include <hip/hip_runtime.h>
#include <hip/hip_bf16.h>
#include <math.h>

// MultiBoxLoss for MI455X (gfx1250), wave32.
// B=64 batches, P=8732 priors, C=21 classes, O=8 truths.
// Pipeline: match -> conf/loc loss -> per-batch bitonic mining -> finalize.
// Reductions use V_WMMA_F32_16X16X4_F32 ones-trick (exact fp32, deterministic).
// conf_data (47 MB, the dominant HBM traffic) is staged into LDS with
// GLOBAL_LOAD_ASYNC_TO_LDS_B128 (ASYNCcnt path) when the toolchain exposes it.

#define BN 64
#define PN 8732
#define CN 21
#define ON 8
#define NBLK_CONF ((BN * PN) / 256)   // 2183 (exact)
#define SORTN 16384                   // next pow2 >= PN; 64 KB LDS

typedef __attribute__((ext_vector_type(2))) float v2f;
typedef __attribute__((ext_vector_type(8))) float v8f;
typedef __attribute__((__vector_size__(4 * sizeof(int)))) int v4i;

#if defined(__gfx1250__) && __has_builtin(__builtin_amdgcn_global_load_async_to_lds_b128)
#define HAVE_ASYNC_LDS 1
#else
#define HAVE_ASYNC_LDS 0
#endif

__device__ __forceinline__ void wait_asynccnt0() {
#if defined(__gfx1250__)
#if __has_builtin(__builtin_amdgcn_s_wait_asynccnt)
  __builtin_amdgcn_s_wait_asynccnt(0);
#else
  asm volatile("s_wait_asynccnt 0x0" ::: "memory");
#endif
#endif
}

// Sum n floats (n % 64 == 0) from LDS using V_WMMA_F32_16X16X4_F32.
// Must be executed by one full wave (EXEC all ones). A = ones(16x4),
// B carries 64 distinct values per iteration; D[m][n] = sum_k B[k][n] + C.
// Summing D row 0 over its 16 columns (xor-shuffles within the 16-group)
// yields the exact total regardless of the (k,n) slot mapping.
__device__ __forceinline__ float wave_wmma_sum(const float* s, int n) {
  const int lane = threadIdx.x & 31;
  v2f a; a.x = 1.0f; a.y = 1.0f;
  v8f c = {0.f, 0.f, 0.f, 0.f, 0.f, 0.f, 0.f, 0.f};
  for (int base = 0; base < n; base += 64) {
    v2f b;
    b.x = s[base + lane];
    b.y = s[base + 32 + lane];
    c = __builtin_amdgcn_wmma_f32_16x16x4_f32(
        /*neg_a=*/false, a, /*neg_b=*/false, b,
        /*c_mod=*/(short)0, c, /*reuse_a=*/false, /*reuse_b=*/false);
  }
  float v = c[0];                 // lanes 0..15: D[0][n], 16..31: D[8][n] (same col sums)
  v += __shfl_xor(v, 1);
  v += __shfl_xor(v, 2);
  v += __shfl_xor(v, 4);
  v += __shfl_xor(v, 8);          // each 16-lane group now holds the full total
  return v;                       // lane 0 valid
}

// ---------------- Kernel A: prior<->truth matching + loc encoding ----------------
__global__ __launch_bounds__(256) void match_kernel(
    const float* __restrict__ priors, const float* __restrict__ targets,
    float* __restrict__ loc_t, int* __restrict__ conf_t, int* __restrict__ nposi) {
  __shared__ float s_t[ON * 5];
  __shared__ float s_area[ON];
  __shared__ int   s_bpi[ON];
  __shared__ float s_rv[256];
  __shared__ int   s_ri[256];
  const int b = blockIdx.x, tid = threadIdx.x;

  if (tid < ON * 5) s_t[tid] = targets[b * ON * 5 + tid];
  __syncthreads();
  if (tid < ON) {
    float x0 = s_t[tid * 5], y0 = s_t[tid * 5 + 1];
    float x1 = s_t[tid * 5 + 2], y1 = s_t[tid * 5 + 3];
    s_area[tid] = (x1 - x0) * (y1 - y0);
  }
  __syncthreads();

  // Pass 1: per-truth argmax over priors.
  float bv[ON]; int bix[ON];
#pragma unroll
  for (int o = 0; o < ON; ++o) { bv[o] = -1.0f; bix[o] = 0x7fffffff; }

  for (int p = tid; p < PN; p += 256) {
    float cx = priors[p * 4], cy = priors[p * 4 + 1];
    float w = priors[p * 4 + 2], h = priors[p * 4 + 3];
    float px0 = cx - 0.5f * w, py0 = cy - 0.5f * h;
    float px1 = cx + 0.5f * w, py1 = cy + 0.5f * h;
    float ab = w * h;
#pragma unroll
    for (int o = 0; o < ON; ++o) {
      float ix0 = fmaxf(s_t[o * 5], px0),     iy0 = fmaxf(s_t[o * 5 + 1], py0);
      float ix1 = fminf(s_t[o * 5 + 2], px1), iy1 = fminf(s_t[o * 5 + 3], py1);
      float iw = fmaxf(ix1 - ix0, 0.f), ih = fmaxf(iy1 - iy0, 0.f);
      float inter = iw * ih;
      float iou = inter / (s_area[o] + ab - inter);
      if (iou > bv[o] || (iou == bv[o] && p < bix[o])) { bv[o] = iou; bix[o] = p; }
    }
  }
  for (int o = 0; o < ON; ++o) {
    s_rv[tid] = bv[o]; s_ri[tid] = bix[o];
    __syncthreads();
    for (int s = 128; s > 0; s >>= 1) {
      if (tid < s) {
        float v2 = s_rv[tid + s]; int i2 = s_ri[tid + s];
        if (v2 > s_rv[tid] || (v2 == s_rv[tid] && i2 < s_ri[tid])) {
          s_rv[tid] = v2; s_ri[tid] = i2;
        }
      }
      __syncthreads();
    }
    if (tid == 0) s_bpi[o] = s_ri[0];
    __syncthreads();
  }

  // Pass 2: per-prior best truth, override with per-truth best prior, encode.
  int npLocal = 0;
  for (int p = tid; p < PN; p += 256) {
    float cx = priors[p * 4], cy = priors[p * 4 + 1];
    float w = priors[p * 4 + 2], h = priors[p * 4 + 3];
    float px0 = cx - 0.5f * w, py0 = cy - 0.5f * h;
    float px1 = cx + 0.5f * w, py1 = cy + 0.5f * h;
    float ab = w * h;
    float bo = -1.0f; int bi = 0;
#pragma unroll
    for (int o = 0; o < ON; ++o) {
      float ix0 = fmaxf(s_t[o * 5], px0),     iy0 = fmaxf(s_t[o * 5 + 1], py0);
      float ix1 = fminf(s_t[o * 5 + 2], px1), iy1 = fminf(s_t[o * 5 + 3], py1);
      float iw = fmaxf(ix1 - ix0, 0.f), ih = fmaxf(iy1 - iy0, 0.f);
      float inter = iw * ih;
      float iou = inter / (s_area[o] + ab - inter);
      if (iou > bo) { bo = iou; bi = o; }
    }
#pragma unroll
    for (int o = 0; o < ON; ++o)
      if (s_bpi[o] == p) { bi = o; bo = 2.0f; }

    int cls = (bo < 0.5f) ? 0 : ((int)s_t[bi * 5 + 4] + 1);
    float mx0 = s_t[bi * 5], my0 = s_t[bi * 5 + 1];
    float mx1 = s_t[bi * 5 + 2], my1 = s_t[bi * 5 + 3];
    size_t idx = (size_t)b * PN + p;
    float gcx = ((mx0 + mx1) * 0.5f - cx) / (0.1f * w);
    float gcy = ((my0 + my1) * 0.5f - cy) / (0.1f * h);
    float gw = logf(fmaxf((mx1 - mx0) / w, 1e-8f)) * 5.0f;   // / VAR1(0.2)
    float gh = logf(fmaxf((my1 - my0) / h, 1e-8f)) * 5.0f;
    loc_t[idx * 4 + 0] = gcx; loc_t[idx * 4 + 1] = gcy;
    loc_t[idx * 4 + 2] = gw;  loc_t[idx * 4 + 3] = gh;
    conf_t[idx] = cls;
    if (cls > 0) ++npLocal;
  }
  s_ri[tid] = npLocal;
  __syncthreads();
  for (int s = 128; s > 0; s >>= 1) {
    if (tid < s) s_ri[tid] += s_ri[tid + s];
    __syncthreads();
  }
  if (tid == 0) nposi[b] = s_ri[0];
}

// --------- Kernel B: conf loss (logsumexp - gathered) + smooth-L1 partials ---------
__global__ __launch_bounds__(256) void conf_loss_kernel(
    const float* __restrict__ conf, const float* __restrict__ loc,
    const float* __restrict__ loc_t, const int* __restrict__ conf_t,
    float* __restrict__ loss_c_all, float* __restrict__ lpart) {
  __shared__ __attribute__((aligned(16))) float s_conf[256 * CN]; // 21504 B tile
  __shared__ float s_red[256];
  const int tid = threadIdx.x;
  const size_t r0 = (size_t)blockIdx.x * 256;
  const float* src = conf + r0 * (size_t)CN;

  __builtin_prefetch(src + 256 * CN, 0, 1);   // global_prefetch_b8: warm next tile

#if HAVE_ASYNC_LDS
  {
    // Async DMA the 21504-byte tile into LDS: 1344 x b128, ASYNCcnt-tracked.
    __attribute__((address_space(1))) v4i* g =
        (__attribute__((address_space(1))) v4i*)src;
    __attribute__((address_space(3))) v4i* l =
        (__attribute__((address_space(3))) v4i*)s_conf;
    for (int i = tid; i < (256 * CN * 4) / 16; i += 256) {
      __builtin_amdgcn_global_load_async_to_lds_b128(g + i, l + i, 0, 0);
    }
    wait_asynccnt0();
  }
#else
  for (int i = tid; i < 256 * CN; i += 256) s_conf[i] = src[i];  // coalesced stage
#endif
  __syncthreads();

  const size_t row = r0 + tid;
  const int ct = conf_t[row];
  const float* cr = &s_conf[tid * CN];   // stride-21: bank-conflict-free
  float m = cr[0];
#pragma unroll
  for (int j = 1; j < CN; ++j) m = fmaxf(m, cr[j]);
  float se = 0.f;
#pragma unroll
  for (int j = 0; j < CN; ++j) se += expf(cr[j] - m);
  float lc = m + logf(se) - cr[ct];
  loss_c_all[row] = lc;

  float part = 0.f;
  if (ct > 0) {
#pragma unroll
    for (int k = 0; k < 4; ++k) {
      float d = loc[row * 4 + k] - loc_t[row * 4 + k];
      float ad = fabsf(d);
      part += (ad < 1.f) ? 0.5f * d * d : (ad - 0.5f);
    }
  }
  s_red[tid] = part;
  __syncthreads();
  if (tid < 32) {
    float t = wave_wmma_sum(s_red, 256);
    if (tid == 0) lpart[blockIdx.x] = t;
  }
}

// --------- Kernel C: per-batch hard-negative mining (bitonic sort in LDS) ---------
__global__ __launch_bounds__(1024) void mine_kernel(
    const float* __restrict__ loss_c_all, const int* __restrict__ conf_t,
    const int* __restrict__ nposi, float* __restrict__ cpart,
    float* __restrict__ npart) {
  __shared__ float s[SORTN];           // exactly 64 KB
  const int b = blockIdx.x, tid = threadIdx.x;
  const int np = nposi[b];
  int nn = 3 * np;
  int lim = PN - np;
  if (nn > lim) nn = lim;
  if (nn < 0) nn = 0;

  float posSum = 0.f;
  const size_t base = (size_t)b * PN;
  for (int i = tid; i < SORTN; i += 1024) {
    float v = -1.0f;                   // pad below any real loss (loss >= 0)
    if (i < PN) {
      float lc = loss_c_all[base + i];
      if (conf_t[base + i] > 0) posSum += lc;   // positives always selected
      else v = lc;                              // mined = where(pos, 0-ish, loss)
    }
    s[i] = v;
  }
  __syncthreads();

  // Bitonic sort, descending. top-k SUM is tie-invariant, so full sort is exact.
  for (int k = 2; k <= SORTN; k <<= 1) {
    for (int j = k >> 1; j > 0; j >>= 1) {
      for (int t = tid; t < SORTN; t += 1024) {
        int ixj = t ^ j;
        if (ixj > t) {
          float a0 = s[t], a1 = s[ixj];
          bool up = ((t & k) == 0);
          bool sw = up ? (a0 < a1) : (a0 > a1);
          if (sw) { s[t] = a1; s[ixj] = a0; }
        }
      }
      __syncthreads();
    }
  }

  float part = posSum;
  for (int i = tid; i < nn; i += 1024) part += s[i];  // top-nn negatives
  __syncthreads();
  s[tid] = part;                      // reuse LDS for the 1024 partials
  __syncthreads();
  if (tid < 32) {
    float t = wave_wmma_sum(s, 1024);
    if (tid == 0) { cpart[b] = t; npart[b] = (float)np; }
  }
}

// ---------------- Kernel D: deterministic final reduction + divide ----------------
__global__ __launch_bounds__(256) void finalize_kernel(
    const float* __restrict__ lpart, const float* __restrict__ cpart,
    const float* __restrict__ npart, float* __restrict__ out) {
  __shared__ float sr[256];
  __shared__ float res[2];
  const int tid = threadIdx.x;

  float a = 0.f;
  for (int i = tid; i < NBLK_CONF; i += 256) a += lpart[i];
  sr[tid] = a;
  __syncthreads();
  if (tid < 32) { float t = wave_wmma_sum(sr, 256); if (tid == 0) res[0] = t; }
  __syncthreads();

  sr[tid] = (tid < BN) ? cpart[tid] : 0.f;
  __syncthreads();
  if (tid < 32) { float t = wave_wmma_sum(sr, 256); if (tid == 0) res[1] = t; }
  __syncthreads();

  sr[tid] = (tid < BN) ? npart[tid] : 0.f;
  __syncthreads();
  if (tid < 32) {
    float n = wave_wmma_sum(sr, 256);
    if (tid == 0) {
      out[0] = res[0] / n;
      out[1] = res[1] / n;
    }
  }
}

extern "C" void kernel_launch(void* const* d_in, const int* in_sizes, int n_in,
                              void* d_out, int out_size, void* d_ws, size_t ws_size,
                              hipStream_t stream) {
  const float* loc     = (const float*)d_in[0];   // (B,P,4)
  const float* conf    = (const float*)d_in[1];   // (B,P,C)
  const float* priors  = (const float*)d_in[2];   // (P,4)
  const float* targets = (const float*)d_in[3];   // (B,O,5)
  float* out = (float*)d_out;

  char* ws = (char*)d_ws;
  size_t off = 0;
  auto align16 = [](size_t x) { return (x + 15) & ~(size_t)15; };

  float* loc_t   = (float*)(ws + off); off = align16(off + (size_t)BN * PN * 4 * sizeof(float));
  int*   conf_t  = (int*)  (ws + off); off = align16(off + (size_t)BN * PN * sizeof(int));
  float* loss_c  = (float*)(ws + off); off = align16(off + (size_t)BN * PN * sizeof(float));
  int*   nposi   = (int*)  (ws + off); off = align16(off + (size_t)BN * sizeof(int));
  float* cpart   = (float*)(ws + off); off = align16(off + (size_t)BN * sizeof(float));
  float* npart   = (float*)(ws + off); off = align16(off + (size_t)BN * sizeof(float));
  float* lpart   = (float*)(ws + off); off = align16(off + (size_t)NBLK_CONF * sizeof(float));

  match_kernel<<<BN, 256, 0, stream>>>(priors, targets, loc_t, conf_t, nposi);
  conf_loss_kernel<<<NBLK_CONF, 256, 0, stream>>>(conf, loc, loc_t, conf_t, loss_c, lpart);
  mine_kernel<<<BN, 1024, 0, stream>>>(loss_c, conf_t, nposi, cpart, npart);
  finalize_kernel<<<1, 256, 0, stream>>>(lpart, cpart, npart, out);
}